// MultiHopMGAT_32461362823666
// MI455X (gfx1250) — compile-verified
//
#include <hip/hip_runtime.h>
#include <hip/hip_bf16.h>
#include <math.h>

#define N_NODES 20000
#define IN_CH   128
#define HID     16
#define HEADS   8
#define OUT_CH  40
#define SLOPE   0.2f

#define KDIM       128     // contraction dim for both layers (IN_CH and HEADS*HID)
#define LDS_STRIDE 132     // 128 + 4 pad: row r starts at bank (4*r)%64 -> conflict-free

typedef __attribute__((ext_vector_type(2))) float v2f;
typedef __attribute__((ext_vector_type(8))) float v8f;

// ---------------------------------------------------------------------------
// fp32 WMMA GEMM: D[M,NCOLS] = A[M,KDIM] * B[KDIM,NCOLS], row major.
// Block = 256 threads = 8 waves, one M-tile per block, one 16x16 N-tile per
// wave. A panel (16x128 fp32 = 8KB) staged once in LDS and shared by all
// waves; B stays in L2 (tiny, reused by all 1250 blocks). K fully unrolled;
// NCOLS is a template constant so B addresses fold into load immediates.
// ISA 7.12.2 32-bit layouts:
//   A 16x4 : lanes 0-15 -> {K=0,K=1}, lanes 16-31 -> {K=2,K=3}, M = lane&15
//   B 4x16 : lanes 0-15 -> {K=0,K=1}, lanes 16-31 -> {K=2,K=3}, N = lane&15
//   C/D    : vgpr r, lane<16 -> (M=r, N=lane); lane>=16 -> (M=r+8, N=lane-16)
// ---------------------------------------------------------------------------
template <int NCOLS>
__global__ void gemm_wmma_f32_kernel(const float* __restrict__ A,
                                     const float* __restrict__ B,
                                     float* __restrict__ D,
                                     int nTilesN) {
    __shared__ float As[16 * LDS_STRIDE];

    const int tileM = blockIdx.x;

    // cooperative, coalesced A-panel load: 2048 floats / 256 threads = 8 each
    {
        const float* __restrict__ Ap = A + (size_t)tileM * 16 * KDIM;
        __builtin_prefetch(Ap, 0, 1);                       // global_prefetch_b8
        for (int i = threadIdx.x; i < 16 * KDIM; i += blockDim.x) {
            const int r = i >> 7;
            const int c = i & (KDIM - 1);
            As[r * LDS_STRIDE + c] = Ap[i];
        }
    }
    __syncthreads();

    const int wave  = threadIdx.x >> 5;
    const int tileN = blockIdx.y * (blockDim.x >> 5) + wave;
    if (tileN >= nTilesN) return;          // whole-wave exit: EXEC all-1s for WMMA below

    const int lane  = threadIdx.x & 31;
    const int half  = lane >> 4;           // 0: K{0,1}, 1: K{2,3}
    const int l     = lane & 15;
    const int col   = tileN * 16 + l;
    const int colc  = (col < NCOLS) ? col : 0;   // clamp + multiply mask (no EXEC games)
    const float bm  = (col < NCOLS) ? 1.0f : 0.0f;

    const float* __restrict__ arow = &As[l * LDS_STRIDE];
    const float* __restrict__ bcol = B + colc;   // single base; k0*NCOLS folds into IOFFSET

    v8f c = {};
#pragma unroll
    for (int kb = 0; kb < KDIM; kb += 4) {
        const int k0 = kb + 2 * half;
        v2f a, b;
        a.x = arow[k0];                    // ds_load_b64, constant offsets after unroll
        a.y = arow[k0 + 1];
        b.x = bcol[k0 * NCOLS] * bm;       // global_load_b32 with immediate offset
        b.y = bcol[(k0 + 1) * NCOLS] * bm;
        c = __builtin_amdgcn_wmma_f32_16x16x4_f32(false, a, false, b,
                                                  (short)0, c, false, false);
    }

#pragma unroll
    for (int r = 0; r < 8; ++r) {
        const int row = tileM * 16 + r + 8 * half;
        if (col < NCOLS) D[(size_t)row * NCOLS + col] = c[r];
    }
}

// ---------------------------------------------------------------------------
// utility kernels
// ---------------------------------------------------------------------------
__global__ void fill_kernel(float* __restrict__ p, float v, int n) {
    int i = blockIdx.x * blockDim.x + threadIdx.x;
    if (i < n) p[i] = v;
}

__global__ void softmax2x2_kernel(const float* __restrict__ hw1,
                                  const float* __restrict__ hw2,
                                  float* __restrict__ hsm) {
    if (threadIdx.x == 0 && blockIdx.x == 0) {
        {
            float a = hw1[0], b = hw1[1];
            float mx = fmaxf(a, b);
            float ea = __expf(a - mx), eb = __expf(b - mx);
            float s = ea + eb;
            hsm[0] = ea / s; hsm[1] = eb / s;
        }
        {
            float a = hw2[0], b = hw2[1];
            float mx = fmaxf(a, b);
            float ea = __expf(a - mx), eb = __expf(b - mx);
            float s = ea + eb;
            hsm[2] = ea / s; hsm[3] = eb / s;
        }
    }
}

// per-node attention dot products: ssrc[n,h] = <h[n,h,:], a_src[h,:]>, same for dst
__global__ void node_scores_kernel(const float* __restrict__ h,
                                   const float* __restrict__ a_src,
                                   const float* __restrict__ a_dst,
                                   float* __restrict__ ssrc,
                                   float* __restrict__ sdst,
                                   int n, int heads, int hid) {
    int i = blockIdx.x * blockDim.x + threadIdx.x;
    if (i >= n * heads) return;
    const int node = i / heads;
    const int hd   = i % heads;
    const float* __restrict__ hp = h + (size_t)node * heads * hid + (size_t)hd * hid;
    const float* __restrict__ as = a_src + (size_t)hd * hid;
    const float* __restrict__ ad = a_dst + (size_t)hd * hid;
    float s1 = 0.0f, s2 = 0.0f;
    for (int c2 = 0; c2 < hid; ++c2) {
        float v = hp[c2];
        s1 += v * as[c2];
        s2 += v * ad[c2];
    }
    ssrc[i] = s1;
    sdst[i] = s2;
}

__device__ __forceinline__ float leaky_relu(float v) {
    return v > 0.0f ? v : SLOPE * v;
}

// ordered float atomic max via signed-int max (>=0) / unsigned-int min (<0)
__device__ __forceinline__ void atomicMaxFloat(float* addr, float val) {
    if (val >= 0.0f) {
        atomicMax((int*)addr, __float_as_int(val));
    } else {
        atomicMin((unsigned int*)addr, __float_as_uint(val));
    }
}

__global__ void edge_max_kernel(const int* __restrict__ src,
                                const int* __restrict__ dst,
                                int E, int heads,
                                const float* __restrict__ ssrc,
                                const float* __restrict__ sdst,
                                float* __restrict__ m) {
    int i = blockIdx.x * blockDim.x + threadIdx.x;
    if (i >= E * heads) return;
    const int e  = i / heads;
    const int hd = i % heads;
    const int is = src[e] * heads + hd;
    const int id = dst[e] * heads + hd;
    atomicMaxFloat(&m[id], leaky_relu(ssrc[is] + sdst[id]));
}

__global__ void edge_expsum_kernel(const int* __restrict__ src,
                                   const int* __restrict__ dst,
                                   int E, int heads,
                                   const float* __restrict__ ssrc,
                                   const float* __restrict__ sdst,
                                   const float* __restrict__ m,
                                   float* __restrict__ z) {
    int i = blockIdx.x * blockDim.x + threadIdx.x;
    if (i >= E * heads) return;
    const int e  = i / heads;
    const int hd = i % heads;
    const int is = src[e] * heads + hd;
    const int id = dst[e] * heads + hd;
    const float v = leaky_relu(ssrc[is] + sdst[id]);
    atomicAdd(&z[id], __expf(v - m[id]));
}

__global__ void edge_msg_kernel(const int* __restrict__ src,
                                const int* __restrict__ dst,
                                int E, int heads, int hid,
                                const float* __restrict__ ssrc,
                                const float* __restrict__ sdst,
                                const float* __restrict__ m,
                                const float* __restrict__ z,
                                const float* __restrict__ h,
                                const float* __restrict__ hw, int k,
                                float* __restrict__ acc) {
    int i = blockIdx.x * blockDim.x + threadIdx.x;
    if (i >= E * heads) return;
    const int e  = i / heads;
    const int hd = i % heads;
    const int s  = src[e];
    const int d  = dst[e];
    const int is = s * heads + hd;
    const int id = d * heads + hd;
    const float v     = leaky_relu(ssrc[is] + sdst[id]);
    const float alpha = __expf(v - m[id]) / (z[id] + 1e-16f);
    const float coef  = hw[k] * alpha;
    const float* __restrict__ hp = h   + (size_t)s * heads * hid + (size_t)hd * hid;
    float* __restrict__       op = acc + (size_t)d * heads * hid + (size_t)hd * hid;
    for (int c2 = 0; c2 < hid; ++c2) {
        atomicAdd(&op[c2], coef * hp[c2]);
    }
}

__global__ void elu_bias_kernel(const float* __restrict__ acc,
                                const float* __restrict__ b,
                                float* __restrict__ out, int total, int C) {
    int i = blockIdx.x * blockDim.x + threadIdx.x;
    if (i >= total) return;
    float v = acc[i] + b[i % C];
    out[i] = v > 0.0f ? v : __expf(v) - 1.0f;
}

__global__ void bias_add_kernel(float* __restrict__ out,
                                const float* __restrict__ b, int total, int C) {
    int i = blockIdx.x * blockDim.x + threadIdx.x;
    if (i >= total) return;
    out[i] += b[i % C];
}

// ---------------------------------------------------------------------------
// launch
// ---------------------------------------------------------------------------
static inline int cdiv(int a, int b) { return (a + b - 1) / b; }

extern "C" void kernel_launch(void* const* d_in, const int* in_sizes, int n_in,
                              void* d_out, int out_size, void* d_ws, size_t ws_size,
                              hipStream_t stream) {
    const float* x   = (const float*)d_in[0];    // [N,128]
    const float* W1  = (const float*)d_in[1];    // [2,128,128]
    const float* as1 = (const float*)d_in[2];    // [2,8,16]
    const float* ad1 = (const float*)d_in[3];    // [2,8,16]
    const float* hw1 = (const float*)d_in[4];    // [2]
    const float* b1  = (const float*)d_in[5];    // [128]
    const float* W2  = (const float*)d_in[6];    // [2,128,40]
    const float* as2 = (const float*)d_in[7];    // [2,1,40]
    const float* ad2 = (const float*)d_in[8];    // [2,1,40]
    const float* hw2 = (const float*)d_in[9];    // [2]
    const float* b2  = (const float*)d_in[10];   // [40]
    const int* e_src[2] = { (const int*)d_in[11], (const int*)d_in[13] };
    const int* e_dst[2] = { (const int*)d_in[12], (const int*)d_in[14] };
    const int  E[2]     = { in_sizes[11], in_sizes[13] };

    const int n  = N_NODES;
    const int HC = HEADS * HID;                  // 128

    // workspace layout (floats)
    float* ws   = (float*)d_ws;
    float* hsm  = ws;                            // 4 (padded to 16)
    float* htmp = ws + 16;                       // n*128 (per-hop GEMM output)
    float* acc1 = htmp + (size_t)n * HC;         // n*128
    float* h1   = acc1 + (size_t)n * HC;         // n*128
    float* ssrc = h1   + (size_t)n * HC;         // n*8
    float* sdst = ssrc + (size_t)n * HEADS;      // n*8
    float* mbuf = sdst + (size_t)n * HEADS;      // n*8
    float* zbuf = mbuf + (size_t)n * HEADS;      // n*8
    float* outf = (float*)d_out;                 // [N,40]

    const int TB = 256;
    const int WAVES = TB / 32;

    softmax2x2_kernel<<<1, 32, 0, stream>>>(hw1, hw2, hsm);

    // ===================== Layer 1: heads=8, hid=16, concat =====================
    fill_kernel<<<cdiv(n * HC, TB), TB, 0, stream>>>(acc1, 0.0f, n * HC);

    for (int k = 0; k < 2; ++k) {
        // h = x @ W1[k]  -> htmp [N,128]
        {
            dim3 grid(n / 16, cdiv(HC / 16, WAVES));
            gemm_wmma_f32_kernel<128><<<grid, TB, 0, stream>>>(
                x, W1 + (size_t)k * IN_CH * HC, htmp, HC / 16);
        }
        node_scores_kernel<<<cdiv(n * HEADS, TB), TB, 0, stream>>>(
            htmp, as1 + (size_t)k * HC, ad1 + (size_t)k * HC,
            ssrc, sdst, n, HEADS, HID);

        fill_kernel<<<cdiv(n * HEADS, TB), TB, 0, stream>>>(mbuf, -INFINITY, n * HEADS);
        fill_kernel<<<cdiv(n * HEADS, TB), TB, 0, stream>>>(zbuf, 0.0f, n * HEADS);

        const int EH = E[k] * HEADS;
        edge_max_kernel<<<cdiv(EH, TB), TB, 0, stream>>>(
            e_src[k], e_dst[k], E[k], HEADS, ssrc, sdst, mbuf);
        edge_expsum_kernel<<<cdiv(EH, TB), TB, 0, stream>>>(
            e_src[k], e_dst[k], E[k], HEADS, ssrc, sdst, mbuf, zbuf);
        edge_msg_kernel<<<cdiv(EH, TB), TB, 0, stream>>>(
            e_src[k], e_dst[k], E[k], HEADS, HID,
            ssrc, sdst, mbuf, zbuf, htmp, hsm, k, acc1);
    }

    // h1 = elu(acc1 + b1)
    elu_bias_kernel<<<cdiv(n * HC, TB), TB, 0, stream>>>(acc1, b1, h1, n * HC, HC);

    // ===================== Layer 2: heads=1, hid=40, mean =====================
    fill_kernel<<<cdiv(n * OUT_CH, TB), TB, 0, stream>>>(outf, 0.0f, n * OUT_CH);

    for (int k = 0; k < 2; ++k) {
        // g = h1 @ W2[k] -> htmp [N,40]
        {
            const int nTilesN = cdiv(OUT_CH, 16);     // 3 (padded, masked)
            dim3 grid(n / 16, cdiv(nTilesN, WAVES));
            gemm_wmma_f32_kernel<OUT_CH><<<grid, TB, 0, stream>>>(
                h1, W2 + (size_t)k * HC * OUT_CH, htmp, nTilesN);
        }
        node_scores_kernel<<<cdiv(n, TB), TB, 0, stream>>>(
            htmp, as2 + (size_t)k * OUT_CH, ad2 + (size_t)k * OUT_CH,
            ssrc, sdst, n, 1, OUT_CH);

        fill_kernel<<<cdiv(n, TB), TB, 0, stream>>>(mbuf, -INFINITY, n);
        fill_kernel<<<cdiv(n, TB), TB, 0, stream>>>(zbuf, 0.0f, n);

        edge_max_kernel<<<cdiv(E[k], TB), TB, 0, stream>>>(
            e_src[k], e_dst[k], E[k], 1, ssrc, sdst, mbuf);
        edge_expsum_kernel<<<cdiv(E[k], TB), TB, 0, stream>>>(
            e_src[k], e_dst[k], E[k], 1, ssrc, sdst, mbuf, zbuf);
        edge_msg_kernel<<<cdiv(E[k], TB), TB, 0, stream>>>(
            e_src[k], e_dst[k], E[k], 1, OUT_CH,
            ssrc, sdst, mbuf, zbuf, htmp, hsm + 2, k, outf);
    }

    bias_add_kernel<<<cdiv(n * OUT_CH, TB), TB, 0, stream>>>(outf, b2, n * OUT_CH, OUT_CH);

    (void)n_in; (void)out_size; (void)ws_size;
}